// Attention_26800595927544
// MI455X (gfx1250) — compile-verified
//
#include <hip/hip_runtime.h>

typedef __attribute__((ext_vector_type(2))) float v2f;
typedef __attribute__((ext_vector_type(8))) float v8f;

#define CIN   256
#define COUT  256
#define BATCH 2
#define HH    48
#define WW    48
#define HWSZ  (HH * WW)          // 2304
#define KS    7
#define PADR  3
#define MAPELEMS (BATCH * COUT * HWSZ)   // one q/k/v map, in floats

// ---------------------------------------------------------------------------
// Kernel 1: fused Q/K/V 1x1-conv GEMM using fp32 WMMA (V_WMMA_F32_16X16X4_F32)
// out[m, n] = sum_k W[m, k] * X[k, n],  M = Cout, K = Cin, N = H*W
// grid: (M/16, N/128, 3*B); block: 256 threads (8 waves, each one 16x16 tile)
// ---------------------------------------------------------------------------
__global__ __launch_bounds__(256) void qkv_gemm(
    const float* __restrict__ x,
    const float* __restrict__ wq,
    const float* __restrict__ wk,
    const float* __restrict__ wv,
    float* __restrict__ ws)
{
  const int z     = blockIdx.z;          // 0..5
  const int b     = z / 3;
  const int which = z - 3 * b;           // 0=q, 1=k, 2=v
  const float* __restrict__ Wm = (which == 0) ? wq : ((which == 1) ? wk : wv);
  const float* __restrict__ X  = x + (size_t)b * CIN * HWSZ;
  float* __restrict__ out = ws + (size_t)which * MAPELEMS + (size_t)b * COUT * HWSZ;

  const int tid  = threadIdx.x;
  const int wave = tid >> 5;
  const int lane = tid & 31;
  const int l15  = lane & 15;
  const int hi   = lane >> 4;            // 0: lanes 0-15, 1: lanes 16-31
  const int koff = hi << 1;              // K offset within a K=4 step

  const int M0 = blockIdx.x * 16;
  const int N0 = blockIdx.y * 128 + wave * 16;
  const int mrow = M0 + l15;             // A: M striped over lanes
  const int ncol = N0 + l15;             // B/C/D: N striped over lanes

  const float* __restrict__ Arow = Wm + (size_t)mrow * CIN;

  v8f acc = {};
  #pragma unroll 8
  for (int k = 0; k < CIN; k += 4) {
    // A[M=mrow, K=k+koff .. k+koff+1]  (contiguous -> 8-byte load)
    v2f a = *reinterpret_cast<const v2f*>(Arow + k + koff);
    // B[K, N=ncol], K rows are HWSZ apart
    v2f bm;
    bm.x = X[(size_t)(k + koff)     * HWSZ + ncol];
    bm.y = X[(size_t)(k + koff + 1) * HWSZ + ncol];
    // (neg_a, A, neg_b, B, c_mod, C, reuse_a, reuse_b)
    acc = __builtin_amdgcn_wmma_f32_16x16x4_f32(
        false, a, false, bm, (short)0, acc, false, false);
  }

  // C/D layout: VGPR r -> M = r (lanes 0-15) or M = 8+r (lanes 16-31), N = lane&15
  const int ms = M0 + (hi << 3);
  #pragma unroll
  for (int r = 0; r < 8; ++r)
    out[(size_t)(ms + r) * HWSZ + ncol] = acc[r];
}

// ---------------------------------------------------------------------------
// Kernel 2: fused local-window attention with LDS-staged K/V patches
// block = 256 threads = 16x16 spatial tile; loops over 16 channels
// grid: (9 spatial tiles, COUT/16 channel groups, B)
// ---------------------------------------------------------------------------
#define TS    16
#define PATCH (TS + KS - 1)    // 22
#define PITCH (PATCH + 1)      // 23 (avoid LDS bank conflicts)
#define CG    16               // channels per block

__global__ __launch_bounds__(256) void local_attn(
    const float* __restrict__ ws,
    const float* __restrict__ rel_h,
    const float* __restrict__ rel_w,
    float* __restrict__ out)
{
  const float* __restrict__ qm = ws;
  const float* __restrict__ km = ws + (size_t)MAPELEMS;
  const float* __restrict__ vm = ws + (size_t)2 * MAPELEMS;

  __shared__ float sk[PATCH * PITCH];
  __shared__ float sv[PATCH * PITCH];

  const int tid = threadIdx.x;
  const int tx  = tid & 15;
  const int ty  = tid >> 4;

  const int b   = blockIdx.z;
  const int c0  = blockIdx.y * CG;
  const int nTw = WW / TS;
  const int th0 = (blockIdx.x / nTw) * TS;
  const int tw0 = (blockIdx.x % nTw) * TS;
  const int h   = th0 + ty;
  const int w   = tw0 + tx;

  for (int ci = 0; ci < CG; ++ci) {
    const int c = c0 + ci;
    const float* __restrict__ kc = km + ((size_t)b * COUT + c) * HWSZ;
    const float* __restrict__ vc = vm + ((size_t)b * COUT + c) * HWSZ;

    // cooperative halo-patch load (zero-fill == reference's zero padding)
    for (int idx = tid; idx < PATCH * PATCH; idx += 256) {
      const int r  = idx / PATCH;
      const int cc = idx - r * PATCH;
      const int gh = th0 - PADR + r;
      const int gw = tw0 - PADR + cc;
      const bool ok = (gh >= 0) & (gh < HH) & (gw >= 0) & (gw < WW);
      sk[r * PITCH + cc] = ok ? kc[gh * WW + gw] : 0.0f;
      sv[r * PITCH + cc] = ok ? vc[gh * WW + gw] : 0.0f;
    }
    __syncthreads();

    const float q = qm[((size_t)b * COUT + c) * HWSZ + h * WW + w];

    // rel bias: first half of channels biased over kh, second over kw
    const bool isrow = (c < COUT / 2);
    const float* __restrict__ rel =
        isrow ? (rel_h + (size_t)c * KS) : (rel_w + (size_t)(c - COUT / 2) * KS);
    float bias7[KS];
    #pragma unroll
    for (int i = 0; i < KS; ++i) bias7[i] = rel[i];   // uniform -> scalar loads

    // pass 1: running max of logits
    float mx = -3.0e38f;
    #pragma unroll
    for (int kh = 0; kh < KS; ++kh) {
      #pragma unroll
      for (int kw = 0; kw < KS; ++kw) {
        const float bias = isrow ? bias7[kh] : bias7[kw];
        const float kv = sk[(ty + kh) * PITCH + (tx + kw)] + bias;
        mx = fmaxf(mx, q * kv);
      }
    }

    // pass 2: exp-sum and weighted V accumulation
    float s = 0.0f, accv = 0.0f;
    #pragma unroll
    for (int kh = 0; kh < KS; ++kh) {
      #pragma unroll
      for (int kw = 0; kw < KS; ++kw) {
        const float bias = isrow ? bias7[kh] : bias7[kw];
        const float kv = sk[(ty + kh) * PITCH + (tx + kw)] + bias;
        const float e  = __expf(q * kv - mx);
        s    += e;
        accv += e * sv[(ty + kh) * PITCH + (tx + kw)];
      }
    }

    out[((size_t)b * COUT + c) * HWSZ + h * WW + w] = accv / s;
    __syncthreads();
  }
}

// ---------------------------------------------------------------------------
extern "C" void kernel_launch(void* const* d_in, const int* in_sizes, int n_in,
                              void* d_out, int out_size, void* d_ws, size_t ws_size,
                              hipStream_t stream) {
  const float* x  = (const float*)d_in[0];
  const float* wq = (const float*)d_in[1];
  const float* wk = (const float*)d_in[2];
  const float* wv = (const float*)d_in[3];
  const float* rh = (const float*)d_in[4];
  const float* rw = (const float*)d_in[5];
  float* ws  = (float*)d_ws;   // [q | k | v] maps, 3 * MAPELEMS floats
  float* out = (float*)d_out;

  dim3 g1(COUT / 16, HWSZ / 128, 3 * BATCH);   // (16, 18, 6)
  qkv_gemm<<<g1, 256, 0, stream>>>(x, wq, wk, wv, ws);

  dim3 g2((HH / TS) * (WW / TS), COUT / CG, BATCH);  // (9, 16, 2)
  local_attn<<<g2, 256, 0, stream>>>(ws, rh, rw, out);
}